// CrossLayer_5385888989703
// MI455X (gfx1250) — compile-verified
//
#include <hip/hip_runtime.h>

typedef float v2f __attribute__((ext_vector_type(2)));
typedef float v8f __attribute__((ext_vector_type(8)));

#define F_DIM   1024
#define L_LAYERS   3

// One 256-thread block (8 wave32) owns a 16-row tile of x.
// Each wave owns a 128-column slab; within a wave, lane L (<16) owns the 64
// contiguous cols [0,64) of row L in that slab, lane L+16 owns cols [64,128).
// The per-row dots are computed with V_WMMA_F32_16X16X4_F32, w replicated
// across B's 16 columns so every D column carries s[row].
__global__ __launch_bounds__(256) void dcn_cross_wmma_kernel(
    const float* __restrict__ x, const float* __restrict__ kernels,
    const float* __restrict__ bias, float* __restrict__ out)
{
    __shared__ float w_lds[L_LAYERS * F_DIM];
    __shared__ float b_lds[L_LAYERS * F_DIM];
    __shared__ float s_partial[8][16];
    __shared__ float s_final[16];

    const int tid     = threadIdx.x;
    const int wave    = tid >> 5;
    const int lane    = tid & 31;
    const int half    = lane >> 4;          // 0 -> k={0,1}, 1 -> k={2,3}
    const int row     = lane & 15;          // matrix row this lane owns
    const int colbase = wave * 128 + half * 64;  // lane's 64-col window
    const int grow    = blockIdx.x * 16 + row;   // global row

    // Load this lane's 64-float window of x (contiguous 256B -> 16x b128).
    float4 xl4[16];
    float4 x04[16];
    const float4* xrow = (const float4*)(x + (size_t)grow * F_DIM + colbase);
#pragma unroll
    for (int j = 0; j < 16; ++j) { xl4[j] = xrow[j]; x04[j] = xl4[j]; }

    // Stage all layers' w and b into LDS (24 KB; L2-resident source).
    for (int i = tid; i < L_LAYERS * F_DIM; i += 256) {
        w_lds[i] = kernels[i];   // kernels: [L, F, 1] flat
        b_lds[i] = bias[i];      // bias:    [L, F, 1] flat
    }
    __syncthreads();

#pragma unroll
    for (int l = 0; l < L_LAYERS; ++l) {
        const float* wl = w_lds + l * F_DIM + colbase;

        // s partials via tensor pipe: 32 chunks of K=4 cover the 128-col slab.
        v8f acc = {};
#pragma unroll
        for (int c = 0; c < 32; ++c) {
            const int j = c >> 1;
            v2f a;
            if (c & 1) { a[0] = xl4[j].z; a[1] = xl4[j].w; }
            else       { a[0] = xl4[j].x; a[1] = xl4[j].y; }
            v2f b = *(const v2f*)(wl + 2 * c);   // broadcast ds_load_b64
            acc = __builtin_amdgcn_wmma_f32_16x16x4_f32(
                false, a, false, b, (short)0, acc, false, false);
        }

        // D col 0: lane 0 has rows 0..7 in acc[0..7], lane 16 has rows 8..15.
        if (lane == 0 || lane == 16) {
            float* dst = &s_partial[wave][half * 8];
#pragma unroll
            for (int m = 0; m < 8; ++m) dst[m] = acc[m];
        }
        __syncthreads();

        if (tid < 16) {
            float s = 0.f;
#pragma unroll
            for (int wv = 0; wv < 8; ++wv) s += s_partial[wv][tid];
            s_final[tid] = s;
        }
        __syncthreads();

        // Update: xl = x0 * s[row] + b + xl   (each lane owns exactly one row)
        const float s = s_final[row];
        const float4* bl = (const float4*)(b_lds + l * F_DIM + colbase);
#pragma unroll
        for (int j = 0; j < 16; ++j) {
            float4 bv = bl[j];
            xl4[j].x = fmaf(x04[j].x, s, bv.x + xl4[j].x);
            xl4[j].y = fmaf(x04[j].y, s, bv.y + xl4[j].y);
            xl4[j].z = fmaf(x04[j].z, s, bv.z + xl4[j].z);
            xl4[j].w = fmaf(x04[j].w, s, bv.w + xl4[j].w);
        }
    }

    // Contiguous 256B store per lane (16x b128).
    float4* orow = (float4*)(out + (size_t)grow * F_DIM + colbase);
#pragma unroll
    for (int j = 0; j < 16; ++j) orow[j] = xl4[j];
}

extern "C" void kernel_launch(void* const* d_in, const int* in_sizes, int n_in,
                              void* d_out, int out_size, void* d_ws, size_t ws_size,
                              hipStream_t stream) {
    const float* x       = (const float*)d_in[0];
    const float* kernels = (const float*)d_in[1];
    const float* bias    = (const float*)d_in[2];
    float* out           = (float*)d_out;

    const int B = in_sizes[0] / F_DIM;   // 16384
    dim3 grid(B / 16), block(256);
    dcn_cross_wmma_kernel<<<grid, block, 0, stream>>>(x, kernels, bias, out);
}